// p_gs_e1_e2_67001489817740
// MI455X (gfx1250) — compile-verified
//
#include <hip/hip_runtime.h>

#define F_DIM  256   // node feature dim (K of GEMM)
#define E1_D   128   // e1 embedding dim (N of GEMM)
#define E2_D   64

typedef float v2f __attribute__((ext_vector_type(2)));
typedef float v8f __attribute__((ext_vector_type(8)));

// ---------------------------------------------------------------------------
// Kernel 1: per 16-node tile, compute e1 = relu(feats @ W_e1 + b_e1) with
// V_WMMA_F32_16X16X4_F32, then immediately contract against the predictor
// head: u1[n] = e1_n . W_p[0:128], v1[n] = e1_n . W_p[128:256].
// One wave (wave32) per 16-node tile; 8 waves / 256-thread block.
// ---------------------------------------------------------------------------
__global__ __launch_bounds__(256) void node_e1_wmma_kernel(
    const float* __restrict__ feats, const float* __restrict__ W_e1,
    const float* __restrict__ b_e1,  const float* __restrict__ W_p,
    float* __restrict__ u1, float* __restrict__ v1, int n_tiles)
{
  const int lane = threadIdx.x & 31;
  const int wave = threadIdx.x >> 5;
  const int tile = blockIdx.x * 8 + wave;
  if (tile >= n_tiles) return;            // wave-uniform: EXEC stays all-ones
  const int m0   = tile * 16;
  const int lo   = lane & 15;             // A: row within tile; B/C: column
  const int h    = lane >> 4;             // lane half selects upper K / upper M

  // A fragment source: lane holds feats[m0+lo][kb + 2h + {0,1}]
  const float* arow = feats + (size_t)(m0 + lo) * F_DIM + 2 * h;

  v8f accu = {0.f,0.f,0.f,0.f,0.f,0.f,0.f,0.f};
  v8f accv = {0.f,0.f,0.f,0.f,0.f,0.f,0.f,0.f};

  for (int nt = 0; nt < E1_D / 16; ++nt) {
    const int ncol = nt * 16 + lo;        // this lane's output column
    const float bias = b_e1[ncol];        // bias is per-column -> uniform over r
    v8f c;
    #pragma unroll
    for (int r = 0; r < 8; ++r) c[r] = bias;

    const float* bcol = W_e1 + ncol;      // B: W_e1[kb + 2h + r][ncol]
    #pragma unroll 4
    for (int kb = 0; kb < F_DIM; kb += 4) {
      v2f a, b;
      a[0] = arow[kb + 0];
      a[1] = arow[kb + 1];
      b[0] = bcol[(size_t)(kb + 2 * h + 0) * E1_D];
      b[1] = bcol[(size_t)(kb + 2 * h + 1) * E1_D];
      c = __builtin_amdgcn_wmma_f32_16x16x4_f32(
              /*neg_a=*/false, a, /*neg_b=*/false, b,
              /*c_mod=*/(short)0, c, /*reuse_a=*/false, /*reuse_b=*/false);
    }

    // relu, then fold this 16-column slab into the u/v head dots.
    const float wa = W_p[ncol];           // W_p[0:128]   (i-side e1 weights)
    const float wb = W_p[E1_D + ncol];    // W_p[128:256] (j-side e1 weights)
    #pragma unroll
    for (int r = 0; r < 8; ++r) {
      float e = c[r] > 0.f ? c[r] : 0.f;
      accu[r] += e * wa;
      accv[r] += e * wb;
    }
  }

  // Reduce over the 16 lanes of each half (columns); masks 1,2,4,8 stay
  // within a 16-lane group on wave32.
  #pragma unroll
  for (int s = 1; s < 16; s <<= 1) {
    #pragma unroll
    for (int r = 0; r < 8; ++r) {
      accu[r] += __shfl_xor(accu[r], s, 32);
      accv[r] += __shfl_xor(accv[r], s, 32);
    }
  }
  if (lo == 0) {                          // lanes 0 and 16: rows 0-7 / 8-15
    #pragma unroll
    for (int r = 0; r < 8; ++r) {
      u1[m0 + 8 * h + r] = accu[r];
      v1[m0 + 8 * h + r] = accv[r];
    }
  }
}

// ---------------------------------------------------------------------------
// Kernel 2: e2 = concat(e2_c, e2_d); add e2 head dots and finalize
//   u[n] = u1[n] + e2_n . W_p[256:384]
//   v[n] = v1[n] + e2_n . W_p[384:512]
// One wave per node (coalesced 32-lane row reads), 8 nodes / block.
// ---------------------------------------------------------------------------
__global__ __launch_bounds__(256) void node_e2_kernel(
    const float* __restrict__ e2_c, const float* __restrict__ e2_d,
    const float* __restrict__ W_p,
    const float* __restrict__ u1, const float* __restrict__ v1,
    float* __restrict__ u, float* __restrict__ v, int n_nodes)
{
  const int lane = threadIdx.x & 31;
  const int wave = threadIdx.x >> 5;
  const int n = blockIdx.x * 8 + wave;
  if (n >= n_nodes) return;
  const float* pc = e2_c + (size_t)n * E2_D;
  const float* pd = e2_d + (size_t)n * E2_D;
  const float* wu = W_p + 2 * E1_D;             // W_p[256:384]
  const float* wv = W_p + 2 * E1_D + 2 * E2_D;  // W_p[384:512]

  float pu = pc[lane] * wu[lane]       + pc[lane + 32] * wu[lane + 32]
           + pd[lane] * wu[64 + lane]  + pd[lane + 32] * wu[96 + lane];
  float pv = pc[lane] * wv[lane]       + pc[lane + 32] * wv[lane + 32]
           + pd[lane] * wv[64 + lane]  + pd[lane + 32] * wv[96 + lane];

  #pragma unroll
  for (int s = 1; s < 32; s <<= 1) {
    pu += __shfl_xor(pu, s, 32);
    pv += __shfl_xor(pv, s, 32);
  }
  if (lane == 0) {
    u[n] = u1[n] + pu;
    v[n] = v1[n] + pv;
  }
}

// ---------------------------------------------------------------------------
// Kernel 3: per edge, score = u[i] + v[j] + b_p. Pos edges then neg edges,
// matching the reference tuple concatenation order.
// ---------------------------------------------------------------------------
__global__ __launch_bounds__(256) void edge_kernel(
    const long long* __restrict__ ei_pos, const long long* __restrict__ ei_neg,
    const float* __restrict__ u, const float* __restrict__ v,
    const float* __restrict__ b_p, float* __restrict__ out,
    int pos_e, int neg_e)
{
  const int t = blockIdx.x * blockDim.x + threadIdx.x;
  if (t >= pos_e + neg_e) return;
  long long i, j;
  if (t < pos_e) {
    i = ei_pos[t];
    j = ei_pos[pos_e + t];
  } else {
    const int s = t - pos_e;
    i = ei_neg[s];
    j = ei_neg[neg_e + s];
  }
  out[t] = u[i] + v[j] + b_p[0];
}

extern "C" void kernel_launch(void* const* d_in, const int* in_sizes, int n_in,
                              void* d_out, int out_size, void* d_ws, size_t ws_size,
                              hipStream_t stream) {
  const float*     feats = (const float*)d_in[0];
  const float*     e2_c  = (const float*)d_in[1];
  const float*     e2_d  = (const float*)d_in[2];
  const long long* ei    = (const long long*)d_in[3];   // int64 [2, POS_E]
  const long long* ein   = (const long long*)d_in[4];   // int64 [2, NEG_E]
  const float*     W_e1  = (const float*)d_in[5];
  const float*     b_e1  = (const float*)d_in[6];
  const float*     W_p   = (const float*)d_in[7];
  const float*     b_p   = (const float*)d_in[8];

  const int n_nodes = in_sizes[0] / F_DIM;  // 100000
  const int pos_e   = in_sizes[3] / 2;      // 250000
  const int neg_e   = in_sizes[4] / 2;      // 250000

  float* u1 = (float*)d_ws;                 // 4 * n_nodes floats of scratch
  float* v1 = u1 + n_nodes;
  float* uu = v1 + n_nodes;
  float* vv = uu + n_nodes;

  const int n_tiles = n_nodes / 16;         // 100000 % 16 == 0

  node_e1_wmma_kernel<<<(n_tiles + 7) / 8, 256, 0, stream>>>(
      feats, W_e1, b_e1, W_p, u1, v1, n_tiles);
  node_e2_kernel<<<(n_nodes + 7) / 8, 256, 0, stream>>>(
      e2_c, e2_d, W_p, u1, v1, uu, vv, n_nodes);
  edge_kernel<<<(pos_e + neg_e + 255) / 256, 256, 0, stream>>>(
      ei, ein, uu, vv, b_p, (float*)d_out, pos_e, neg_e);
}